// SimpleInteractionModel_52450140618894
// MI455X (gfx1250) — compile-verified
//
#include <hip/hip_runtime.h>
#include <hip/hip_bf16.h>

#define N_NODES   100000
#define N_EDGES   1600000
#define IN_DIM    128
#define HID       128
#define OUT_DIM   64
#define NLP_DIM   786
#define N_LAYERS  3
#define N_USERS   4096
#define ZDIM      (OUT_DIM + NLP_DIM)   /* 850 */
#define ZPAD      852                    /* padded to multiple of 4 for WMMA K-loop */
#define BN_EPS    1e-5f

typedef float v2f __attribute__((ext_vector_type(2)));
typedef float v8f __attribute__((ext_vector_type(8)));

// ---------------------------------------------------------------------------
// Elementwise / scatter kernels (bandwidth path; b128 loads, atomics in L2)
// ---------------------------------------------------------------------------

__global__ void k_zero(float* __restrict__ p, int n) {
    int i = blockIdx.x * blockDim.x + threadIdx.x;
    if (i < n) p[i] = 0.0f;
}

__global__ void k_count_deg(const int* __restrict__ dst, float* __restrict__ deg, int E) {
    int i = blockIdx.x * blockDim.x + threadIdx.x;
    if (i < E) atomicAdd(&deg[dst[i]], 1.0f);
}

__global__ void k_finalize_dinv(float* __restrict__ deg, int n) {
    int i = blockIdx.x * blockDim.x + threadIdx.x;
    if (i < n) deg[i] = rsqrtf(deg[i] + 1.0f);   // in-place: deg -> dinv
}

// agg[v][c] = dinv[v]^2 * hw[v][c] + b_gnn[c], float4-vectorized (b128)
__global__ void k_self_init(const float4* __restrict__ hw, const float* __restrict__ dinv,
                            const float4* __restrict__ bg4, float4* __restrict__ agg,
                            int total4) {
    int i = blockIdx.x * blockDim.x + threadIdx.x;
    if (i >= total4) return;
    int v  = i >> 5;           // / (HID/4)
    int c4 = i & 31;
    float di = dinv[v];
    float s  = di * di;
    float4 h = hw[i];
    float4 b = bg4[c4];
    agg[i] = make_float4(s * h.x + b.x, s * h.y + b.y, s * h.z + b.z, s * h.w + b.w);
}

// One wave per edge: each lane owns 4 contiguous channels -> single b128 load
// of hw[src], then 4x global_atomic_add_f32 into agg[dst]. Wave covers the
// full 128-channel row contiguously (coalesced 512B).
__global__ void k_scatter(const int* __restrict__ src, const int* __restrict__ dst,
                          const float* __restrict__ dinv, const float* __restrict__ hw,
                          float* __restrict__ agg, int E) {
    int wave = (blockIdx.x * blockDim.x + threadIdx.x) >> 5;
    int lane = threadIdx.x & 31;
    if (wave >= E) return;
    int s = src[wave], d = dst[wave];
    float w = dinv[s] * dinv[d];
    float4 v = ((const float4*)(hw + (size_t)s * HID))[lane];
    float* ad = agg + (size_t)d * HID + lane * 4;
    atomicAdd(ad + 0, w * v.x);
    atomicAdd(ad + 1, w * v.y);
    atomicAdd(ad + 2, w * v.z);
    atomicAdd(ad + 3, w * v.w);
}

// h = relu(batchnorm(agg)), float4-vectorized
__global__ void k_bn_relu(const float4* __restrict__ agg, const float4* __restrict__ gamma4,
                          const float4* __restrict__ beta4, const float4* __restrict__ mean4,
                          const float4* __restrict__ var4, float4* __restrict__ h, int total4) {
    int i = blockIdx.x * blockDim.x + threadIdx.x;
    if (i >= total4) return;
    int c4 = i & 31;
    float4 a = agg[i], g = gamma4[c4], bt = beta4[c4], mn = mean4[c4], vr = var4[c4];
    float4 o;
    o.x = fmaxf((a.x - mn.x) * rsqrtf(vr.x + BN_EPS) * g.x + bt.x, 0.0f);
    o.y = fmaxf((a.y - mn.y) * rsqrtf(vr.y + BN_EPS) * g.y + bt.y, 0.0f);
    o.z = fmaxf((a.z - mn.z) * rsqrtf(vr.z + BN_EPS) * g.z + bt.z, 0.0f);
    o.w = fmaxf((a.w - mn.w) * rsqrtf(vr.w + BN_EPS) * g.w + bt.w, 0.0f);
    h[i] = o;
}

// Z[u] = concat(emb[user_indices[u]], nlp, zero-pad to 852)  (tiny: 3.5M elems)
__global__ void k_build_z(const float* __restrict__ emb, const int* __restrict__ uidx,
                          const float* __restrict__ nlp, float* __restrict__ Z) {
    int i = blockIdx.x * blockDim.x + threadIdx.x;
    if (i >= N_USERS * ZPAD) return;
    int u = i / ZPAD, c = i % ZPAD;
    float v;
    if (c < OUT_DIM)      v = emb[(size_t)uidx[u] * OUT_DIM + c];
    else if (c < ZDIM)    v = nlp[c - OUT_DIM];
    else                  v = 0.0f;
    Z[i] = v;
}

// W1p[852][256] = W1[850][256] zero-padded in K
__global__ void k_pad_w1(const float* __restrict__ W1, float* __restrict__ W1p) {
    int i = blockIdx.x * blockDim.x + threadIdx.x;
    if (i >= ZPAD * 256) return;
    int r = i / 256, c = i % 256;
    W1p[i] = (r < ZDIM) ? W1[r * 256 + c] : 0.0f;
}

// out[u] = sigmoid(dot(z2[u], W3) + b3); one wave per row, b128 loads + shfl
__global__ void k_final_dot(const float* __restrict__ z2, const float* __restrict__ W3,
                            const float* __restrict__ b3, float* __restrict__ out, int B) {
    int wave = (blockIdx.x * blockDim.x + threadIdx.x) >> 5;
    int lane = threadIdx.x & 31;
    if (wave >= B) return;
    float4 r  = ((const float4*)(z2 + (size_t)wave * 128))[lane];
    float4 w4 = ((const float4*)W3)[lane];
    float s = r.x * w4.x + r.y * w4.y + r.z * w4.z + r.w * w4.w;
#pragma unroll
    for (int off = 16; off > 0; off >>= 1) s += __shfl_down(s, off, 32);
    if (lane == 0) out[wave] = 1.0f / (1.0f + expf(-(s + b3[0])));
}

// ---------------------------------------------------------------------------
// WMMA f32 GEMM: C[M,N] = relu?(A[M,K] @ B[K,N] + bias), exact f32 math via
// V_WMMA_F32_16X16X4_F32. One wave per 16x16 output tile; M, N multiples of
// 16; K multiple of 4. Per ISA VGPR layouts:
//   A 16x4 : lane -> m = lane%16, VGPR0 = A[m][2*(lane/16)],   VGPR1 = +1
//   B 4x16 : lane -> n = lane%16, VGPR0 = B[2*(lane/16)][n],   VGPR1 = +1
//   C 16x16: lane -> n = lane%16, VGPR v holds row v + 8*(lane/16)
// Guard is wave-uniform so EXEC stays all-1s for WMMA.
// ---------------------------------------------------------------------------
__global__ void k_gemm_wmma_f32(const float* __restrict__ A, const float* __restrict__ B,
                                const float* __restrict__ bias, float* __restrict__ C,
                                int M, int K, int N, int relu) {
    int wave = (blockIdx.x * blockDim.x + threadIdx.x) >> 5;
    int lane = threadIdx.x & 31;
    int colTiles = N >> 4;
    int totalWaves = (M >> 4) * colTiles;
    if (wave >= totalWaves) return;
    int rowTile = wave / colTiles;
    int colTile = wave % colTiles;

    int m     = lane & 15;
    int khalf = (lane >> 4) << 1;       // 0 or 2
    const float* Arow = A + (size_t)(rowTile * 16 + m) * K;
    const float* Bcol = B + colTile * 16 + (lane & 15);

    v8f acc = {};
#pragma unroll 4
    for (int k = 0; k < K; k += 4) {
        v2f a, b;
        a.x = Arow[k + khalf];
        a.y = Arow[k + khalf + 1];
        b.x = Bcol[(size_t)(k + khalf) * N];
        b.y = Bcol[(size_t)(k + khalf + 1) * N];
        acc = __builtin_amdgcn_wmma_f32_16x16x4_f32(
            /*neg_a=*/false, a, /*neg_b=*/false, b,
            /*c_mod=*/(short)0, acc, /*reuse_a=*/false, /*reuse_b=*/false);
    }

    int n  = colTile * 16 + (lane & 15);
    float bv = bias ? bias[n] : 0.0f;
    int rhi = (lane >> 4) ? 8 : 0;
#pragma unroll
    for (int v = 0; v < 8; ++v) {
        float val = acc[v] + bv;
        if (relu) val = fmaxf(val, 0.0f);
        C[(size_t)(rowTile * 16 + v + rhi) * N + n] = val;
    }
}

// ---------------------------------------------------------------------------
// Host launcher
// ---------------------------------------------------------------------------
static inline void launch_gemm(const float* A, const float* B, const float* bias,
                               float* C, int M, int K, int N, int relu, hipStream_t s) {
    int waves = (M >> 4) * (N >> 4);
    int blocks = (waves + 7) / 8;                 // 8 waves (256 threads) per block
    k_gemm_wmma_f32<<<blocks, 256, 0, s>>>(A, B, bias, C, M, K, N, relu);
}

extern "C" void kernel_launch(void* const* d_in, const int* in_sizes, int n_in,
                              void* d_out, int out_size, void* d_ws, size_t ws_size,
                              hipStream_t stream) {
    const float* x        = (const float*)d_in[0];
    const float* nlp      = (const float*)d_in[1];
    const int*   eidx     = (const int*)  d_in[2];   // [2, E]
    const int*   uidx     = (const int*)  d_in[3];
    const float* W_in     = (const float*)d_in[4];
    const float* b_in     = (const float*)d_in[5];
    const float* W_gnn    = (const float*)d_in[6];   // [3,128,128]
    const float* b_gnn    = (const float*)d_in[7];   // [3,128]
    const float* bn_gamma = (const float*)d_in[8];
    const float* bn_beta  = (const float*)d_in[9];
    const float* bn_mean  = (const float*)d_in[10];
    const float* bn_var   = (const float*)d_in[11];
    const float* W_proj   = (const float*)d_in[12];
    const float* b_proj   = (const float*)d_in[13];
    const float* W1       = (const float*)d_in[14];  // [850,256]
    const float* b1       = (const float*)d_in[15];
    const float* W2       = (const float*)d_in[16];  // [256,128]
    const float* b2       = (const float*)d_in[17];
    const float* W3       = (const float*)d_in[18];  // [128,1]
    const float* b3       = (const float*)d_in[19];
    float* out = (float*)d_out;

    const int* src = eidx;
    const int* dst = eidx + N_EDGES;

    // ---- workspace layout (floats; all big blocks 16B-aligned) ----
    float* ws = (float*)d_ws;
    float* dinv = ws;                              // [N]            (deg -> dinv in place)
    float* hbuf = dinv + N_NODES;                  // [N,128]
    float* hw   = hbuf + (size_t)N_NODES * HID;    // [N,128]
    float* agg  = hw   + (size_t)N_NODES * HID;    // [N,128]
    // reuse after GCN stack:
    float* emb  = hw;                              // [N,64]   (<= N*128)
    float* Zp   = agg;                             // [4096,852]
    float* W1p  = hbuf;                            // [852,256]
    float* z1   = W1p + (size_t)ZPAD * 256;        // [4096,256]
    float* z2   = z1  + (size_t)N_USERS * 256;     // [4096,128]

    const int TB  = 256;
    const int NH4 = N_NODES * HID / 4;             // float4 count

    // 1) degree -> dinv
    k_zero<<<(N_NODES + TB - 1) / TB, TB, 0, stream>>>(dinv, N_NODES);
    k_count_deg<<<(N_EDGES + TB - 1) / TB, TB, 0, stream>>>(dst, dinv, N_EDGES);
    k_finalize_dinv<<<(N_NODES + TB - 1) / TB, TB, 0, stream>>>(dinv, N_NODES);

    // 2) h = relu(x @ W_in + b_in)
    launch_gemm(x, W_in, b_in, hbuf, N_NODES, IN_DIM, HID, 1, stream);

    // 3) GCN layers
    for (int l = 0; l < N_LAYERS; ++l) {
        launch_gemm(hbuf, W_gnn + (size_t)l * HID * HID, nullptr, hw,
                    N_NODES, HID, HID, 0, stream);
        k_self_init<<<(NH4 + TB - 1) / TB, TB, 0, stream>>>(
            (const float4*)hw, dinv, (const float4*)(b_gnn + l * HID),
            (float4*)agg, NH4);
        {
            long long threads = (long long)N_EDGES * 32;
            int blocks = (int)((threads + TB - 1) / TB);
            k_scatter<<<blocks, TB, 0, stream>>>(src, dst, dinv, hw, agg, N_EDGES);
        }
        k_bn_relu<<<(NH4 + TB - 1) / TB, TB, 0, stream>>>(
            (const float4*)agg,
            (const float4*)(bn_gamma + l * HID), (const float4*)(bn_beta + l * HID),
            (const float4*)(bn_mean + l * HID),  (const float4*)(bn_var + l * HID),
            (float4*)hbuf, NH4);
    }

    // 4) emb = h @ W_proj + b_proj
    launch_gemm(hbuf, W_proj, b_proj, emb, N_NODES, HID, OUT_DIM, 0, stream);

    // 5) predictor MLP on 4096 users
    k_pad_w1<<<(ZPAD * 256 + TB - 1) / TB, TB, 0, stream>>>(W1, W1p);
    k_build_z<<<(N_USERS * ZPAD + TB - 1) / TB, TB, 0, stream>>>(emb, uidx, nlp, Zp);
    launch_gemm(Zp, W1p, b1, z1, N_USERS, ZPAD, 256, 1, stream);
    launch_gemm(z1, W2, b2, z2, N_USERS, 256, 128, 1, stream);
    {
        int blocks = (N_USERS * 32 + TB - 1) / TB;
        k_final_dot<<<blocks, TB, 0, stream>>>(z2, W3, b3, out, N_USERS);
    }
}